// mixgb_87582973100634
// MI455X (gfx1250) — compile-verified
//
#include <hip/hip_runtime.h>
#include <stdint.h>

// Fused 2x2/stride-2 max-pool + avg-pool + depthwise-2x2 gate convs.
// B=16, C=128, H=W=256 -> out 16x128x128x128 (fp32).
//
// Memory-bound stream (671MB total, ~29us floor at 23.3 TB/s). CDNA5
// async-to-LDS engine with a 3-deep per-wave pipeline: two 2KB tiles
// (8 GLOBAL_LOAD_ASYNC_TO_LDS_B128 transfers) are in flight at all times
// while the wave computes a third from LDS -> maximal memory-level
// parallelism per wave. NT cache policy on both the async loads and the
// output stores keeps the 512MB single-pass stream from rinsing the
// 192MB L2.

typedef __attribute__((ext_vector_type(4))) float v4f;

#define C_  128
#define H_  256
#define W_  256
#define H2  128
#define W2  128

#define WAVES_PER_BLOCK 8
#define CHUNKS_PER_WAVE 16   // 8 waves * 16 rows = 128 output rows = one plane

// Issue one wave-tile: 4 async b128 loads (2 input rows x 32B per lane).
// The instruction's immediate offset is applied to BOTH the LDS and global
// address (CDNA5 ISA); the LDS tile layout mirrors the
// global byte deltas (0, 16, 1024, 1040), so one LDS-address VGPR and one
// 64-bit global-address VGPR cover all four transfers.
__device__ __forceinline__ void issue_async_tile(unsigned lds_addr,
                                                 unsigned long long gaddr) {
    asm volatile(
        "global_load_async_to_lds_b128 %0, %1, off th:TH_LOAD_NT\n\t"
        "global_load_async_to_lds_b128 %0, %1, off offset:16 th:TH_LOAD_NT\n\t"
        "global_load_async_to_lds_b128 %0, %1, off offset:1024 th:TH_LOAD_NT\n\t"
        "global_load_async_to_lds_b128 %0, %1, off offset:1040 th:TH_LOAD_NT"
        :
        : "v"(lds_addr), "v"(gaddr)
        : "memory");
}

__global__ __launch_bounds__(256) void mixgb_fused_async_kernel(
    const float* __restrict__ x,        // (B,C,256,256)
    const float* __restrict__ maxgate,  // (C,1,2,2)
    const float* __restrict__ avggate,  // (C,1,2,2)
    const float* __restrict__ mb,       // (C,)
    const float* __restrict__ ab,       // (C,)
    float* __restrict__ out)            // (B,C,128,128)
{
    // Per-wave triple buffer: 2 input rows (2 x 1024B) per tile, 3 tiles.
    // 8 waves * 3 * 2048B = 48KB LDS.
    __shared__ v4f tiles[WAVES_PER_BLOCK][3][128];

    const int plane = blockIdx.x;          // b*C + c, one plane per block
    const int c     = plane & (C_ - 1);
    const int wave  = threadIdx.x >> 5;
    const int lane  = threadIdx.x & 31;

    // Per-channel parameters: uniform addresses -> scalar loads.
    const float mg0 = maxgate[c * 4 + 0];
    const float mg1 = maxgate[c * 4 + 1];
    const float mg2 = maxgate[c * 4 + 2];
    const float mg3 = maxgate[c * 4 + 3];
    const float ag0 = avggate[c * 4 + 0];
    const float ag1 = avggate[c * 4 + 1];
    const float ag2 = avggate[c * 4 + 2];
    const float ag3 = avggate[c * 4 + 3];
    const float mbias = mb[c];
    const float abias = ab[c];

    // Wave handles output rows r = wave + 8*i, i = 0..15.
    // Global byte address of (plane, input row 2r, col 8*lane):
    //   plane*262144 + r*2048 + lane*32 ; +16384 bytes per chunk step.
    const unsigned long long gbase =
        (unsigned long long)(const char*)x +
        (unsigned long long)plane * (H_ * W_ * 4) +
        (unsigned)(wave * 2048 + lane * 32);

    // LDS byte address for the async VDST operand: low 32 bits of the flat
    // pointer are the LDS offset (flat->LDS mapping truncates to addr[31:0],
    // same LDS_BASE-relative convention as DS ops).
    const unsigned lbase =
        (unsigned)(uintptr_t)&tiles[wave][0][0] + (unsigned)(lane * 32);

    float* obase = out + plane * (H2 * W2) + wave * W2 + lane * 4;

    // Compute-and-store for tile i out of LDS buffer bi (wave-private data:
    // s_wait_asynccnt is sufficient, no barrier needed).
    auto compute_tile = [&](int i, int bi) {
        const v4f* sb = &tiles[wave][bi][0];
        v4f r0a = sb[2 * lane + 0];       // input row 2r,   cols 8l..8l+3
        v4f r0b = sb[2 * lane + 1];       // input row 2r,   cols 8l+4..8l+7
        v4f r1a = sb[64 + 2 * lane + 0];  // input row 2r+1, cols 8l..8l+3
        v4f r1b = sb[64 + 2 * lane + 1];  // input row 2r+1, cols 8l+4..8l+7

        v4f res;
#pragma unroll
        for (int j = 0; j < 4; ++j) {
            float a00, a01, a10, a11;
            if (j < 2) {
                a00 = r0a[2 * j];         a01 = r0a[2 * j + 1];
                a10 = r1a[2 * j];         a11 = r1a[2 * j + 1];
            } else {
                a00 = r0b[2 * (j - 2)];   a01 = r0b[2 * (j - 2) + 1];
                a10 = r1b[2 * (j - 2)];   a11 = r1b[2 * (j - 2) + 1];
            }
            float mx = fmaxf(fmaxf(a00, a01), fmaxf(a10, a11));
            float av = (a00 + a01 + a10 + a11) * 0.25f;
            float mw = fmaf(a00, mg0, fmaf(a01, mg1, fmaf(a10, mg2, fmaf(a11, mg3, mbias))));
            float aw = fmaf(a00, ag0, fmaf(a01, ag1, fmaf(a10, ag2, fmaf(a11, ag3, abias))));
            res[j] = fmaf(mx, mw, av * aw);
        }
        // Output row r = wave + 8*i -> +1024 floats per chunk. NT b128 store.
        __builtin_nontemporal_store(res, (v4f*)(obase + i * (8 * W2)));
    };

    // Prologue: start tiles 0 and 1.
    issue_async_tile(lbase,                 gbase);
    issue_async_tile(lbase + 2048u,         gbase + 16384u);

    // Steady state: issue tile i+2, wait until only tiles i+1,i+2 (8
    // transfers) remain outstanding -> tile i resident (in-order completion).
    int bi2 = 2;  // buffer index of tile i+2 (= (i+2) % 3)
    for (int i = 0; i < CHUNKS_PER_WAVE - 2; ++i) {
        issue_async_tile(lbase + (unsigned)(bi2 * 2048),
                         gbase + (unsigned)((i + 2) * 16384));
        asm volatile("s_wait_asynccnt 0x8" ::: "memory");
        compute_tile(i, bi2 == 2 ? 0 : bi2 + 1);  // bi(i) = (i) % 3
        bi2 = (bi2 == 2) ? 0 : bi2 + 1;
    }

    // Epilogue: tiles 14 (buf 2) and 15 (buf 0).
    asm volatile("s_wait_asynccnt 0x4" ::: "memory");
    compute_tile(CHUNKS_PER_WAVE - 2, (CHUNKS_PER_WAVE - 2) % 3);
    asm volatile("s_wait_asynccnt 0x0" ::: "memory");
    compute_tile(CHUNKS_PER_WAVE - 1, (CHUNKS_PER_WAVE - 1) % 3);
}

extern "C" void kernel_launch(void* const* d_in, const int* in_sizes, int n_in,
                              void* d_out, int out_size, void* d_ws, size_t ws_size,
                              hipStream_t stream) {
    (void)in_sizes; (void)n_in; (void)d_ws; (void)ws_size; (void)out_size;

    const float* x       = (const float*)d_in[0];   // (16,128,256,256)
    const float* maxgate = (const float*)d_in[1];   // (128,1,2,2)
    const float* avggate = (const float*)d_in[2];   // (128,1,2,2)
    const float* mb      = (const float*)d_in[3];   // (128,)
    const float* ab      = (const float*)d_in[4];   // (128,)
    float* out           = (float*)d_out;           // (16,128,128,128)

    const int blocks = 16 * C_;   // 2048 planes, one per block
    mixgb_fused_async_kernel<<<dim3(blocks), dim3(256), 0, stream>>>(
        x, maxgate, avggate, mb, ab, out);
}